// KAN_84421877170604
// MI455X (gfx1250) — compile-verified
//
#include <hip/hip_runtime.h>
#include <hip/hip_bf16.h>

// QSP expectation evaluation, MI455X (gfx1250, wave32).
//
// Per element: theta = x0*x1; v = (e^{i phi0}, 0); 20x v <- W(theta)*Rz(phi_j)*v;
// out = alpha * Re(v0).  Only the first column of U is tracked (4 floats of state).
//
// Data path: raw params staged global->LDS via CDNA5 async op (ASYNCcnt), then each
// wave computes sincos once (lane j owns phi_j) and broadcasts all 21 (c,s) pairs
// into SGPRs via v_readlane_b32 -> the 20-step inner chain is pure VALU, no LDS.

#define QSP_NPARAMS 21   // 2*depth + 1, depth = 10
#define TPB 256          // 8 wave32 per block
#define EPT 4            // elements per thread

typedef __attribute__((address_space(1))) int as1_int;   // global
typedef __attribute__((address_space(3))) int as3_int;   // LDS

__launch_bounds__(TPB)
__global__ void qsp_eval_kernel(const float* __restrict__ x,
                                const float* __restrict__ params,
                                const float* __restrict__ alphas,
                                float* __restrict__ out,
                                int B)
{
    __shared__ float s_raw[QSP_NPARAMS];

    const int tid = threadIdx.x;

    // ---- Stage the 21 shared params into LDS (CDNA5 async global->LDS path) ----
#if defined(__gfx1250__) && __has_builtin(__builtin_amdgcn_global_load_async_to_lds_b32)
    if (tid < QSP_NPARAMS) {
        __builtin_amdgcn_global_load_async_to_lds_b32(
            (as1_int*)(params + tid),
            (as3_int*)(&s_raw[tid]),
            /*offset=*/0, /*cpol=*/0);
    }
  #if __has_builtin(__builtin_amdgcn_s_wait_asynccnt)
    __builtin_amdgcn_s_wait_asynccnt(0);
  #else
    asm volatile("s_wait_asynccnt 0" ::: "memory");
  #endif
#else
    if (tid < QSP_NPARAMS) s_raw[tid] = params[tid];
#endif
    __syncthreads();

    // ---- Per wave: lane j computes (cos phi_j, sin phi_j); broadcast to SGPRs ----
    const int lane = tid & 31;                       // wave32
    const int jj   = (lane < QSP_NPARAMS) ? lane : 0; // keep all lanes defined
    float mys, myc;
    __sincosf(s_raw[jj], &mys, &myc);

    float cj[QSP_NPARAMS], sj[QSP_NPARAMS];
#pragma unroll
    for (int j = 0; j < QSP_NPARAMS; ++j) {
        cj[j] = __int_as_float(__builtin_amdgcn_readlane(__float_as_int(myc), j));
        sj[j] = __int_as_float(__builtin_amdgcn_readlane(__float_as_int(mys), j));
    }

    const float2* __restrict__ x2 = (const float2*)x;  // x is [B,2] interleaved
    const int base0 = blockIdx.x * (TPB * EPT);

#pragma unroll
    for (int e = 0; e < EPT; ++e) {
        const int i = base0 + e * TPB + tid;   // coalesced across the wave
        if (i < B) {
            const float2 xv = x2[i];
            const float  al = alphas[i];

            float st, ct;
            __sincosf(xv.x * xv.y, &st, &ct);   // theta = x0*x1

            // v = first column of U0 = (e^{i phi0}, 0)
            float ar = cj[0], ai = sj[0];
            float br = 0.0f,  bi = 0.0f;

#pragma unroll
            for (int j = 1; j < QSP_NPARAMS; ++j) {
                const float p = cj[j], q = sj[j];
                // Rz(phi): a *= e^{i phi}; b *= e^{-i phi}
                const float tar = __builtin_fmaf(ar, p, -(ai * q));
                const float tai = __builtin_fmaf(ar, q,  (ai * p));
                const float tbr = __builtin_fmaf(br, p,  (bi * q));
                const float tbi = __builtin_fmaf(bi, p, -(br * q));
                // W(theta): (a,b) <- (c*a + i*s*b, i*s*a + c*b)
                ar = __builtin_fmaf(ct, tar, -(st * tbi));
                ai = __builtin_fmaf(ct, tai,  (st * tbr));
                br = __builtin_fmaf(ct, tbr, -(st * tai));
                bi = __builtin_fmaf(ct, tbi,  (st * tar));
            }

            out[i] = al * ar;   // alpha * Re(U00)
        }
    }
}

extern "C" void kernel_launch(void* const* d_in, const int* in_sizes, int n_in,
                              void* d_out, int out_size, void* d_ws, size_t ws_size,
                              hipStream_t stream)
{
    (void)n_in; (void)d_ws; (void)ws_size; (void)out_size;

    const float* x      = (const float*)d_in[0];  // [B,2] f32
    const float* params = (const float*)d_in[1];  // [21]  f32
    const float* alphas = (const float*)d_in[2];  // [B]   f32
    float*       out    = (float*)d_out;          // [B,1] f32

    const int B = in_sizes[2];                    // batch from alphas
    const int per_block = TPB * EPT;
    const int grid = (B + per_block - 1) / per_block;

    qsp_eval_kernel<<<grid, TPB, 0, stream>>>(x, params, alphas, out, B);
}